// BasicGCNNetwork_76046690943380
// MI455X (gfx1250) — compile-verified
//
#include <hip/hip_runtime.h>
#include <hip/hip_bf16.h>

// -------- CDNA5 WMMA types --------
typedef float v2f __attribute__((ext_vector_type(2)));
typedef float v8f __attribute__((ext_vector_type(8)));
typedef int   v4i __attribute__((ext_vector_type(4)));

#define HID 64
#define LDS_KMAX 128
#define LDS_PAD 4   // pad rows so A reads (stride K+4) spread across LDS banks

#if defined(__AMDGCN__) && __has_builtin(__builtin_amdgcn_global_load_async_to_lds_b128)
#define USE_ASYNC_LDS 1
#else
#define USE_ASYNC_LDS 0
#endif

// ---------------- degree / normalization ----------------
__global__ void init_deg_kernel(float* deg, int n) {
    int i = blockIdx.x * blockDim.x + threadIdx.x;
    if (i < n) deg[i] = 1.0f;   // self-loop contributes 1 to every dst degree
}

__global__ void accum_deg_kernel(const int* __restrict__ dst, float* deg, int e) {
    int i = blockIdx.x * blockDim.x + threadIdx.x;
    if (i < e) atomicAdd(&deg[dst[i]], 1.0f);
}

__global__ void rsqrt_deg_kernel(float* deg, int n) {
    int i = blockIdx.x * blockDim.x + threadIdx.x;
    if (i < n) {
        float d = deg[i];
        deg[i] = (d > 0.0f) ? rsqrtf(d) : 0.0f;   // now holds dis = deg^{-1/2}
    }
}

// ---------------- f32 WMMA GEMM: C[M,64] = A[M,K] @ W[K,64] ----------------
// Block = 128 threads (4 waves). Block handles a 16-row stripe; wave w owns
// column tile w. The 16xK A stripe (contiguous in global, row-major) is staged
// once into LDS (async global->LDS on gfx1250 when available), then all 4
// waves feed V_WMMA_F32_16X16X4_F32 from LDS. No divergence: EXEC all-1s.
__global__ void gemm_wmma_f32(const float* __restrict__ A,
                              const float* __restrict__ W,
                              float* __restrict__ C, int K) {
    __shared__ float lds_a[16 * (LDS_KMAX + LDS_PAD)];

    const int tid   = threadIdx.x;
    const int lane  = tid & 31;
    const int wave  = tid >> 5;                  // 0..3 -> column tile
    const int tm    = blockIdx.x;                // row tile (16 rows)
    const int half  = lane >> 4;                 // 0: K pair {0,1}, 1: {2,3}
    const int khalf = half << 1;
    const int m     = lane & 15;
    const int n     = wave * 16 + (lane & 15);   // B/C/D column for this lane
    const int ldsK  = K + LDS_PAD;

    // ---- stage the 16xK A stripe into LDS (16B chunks) ----
    const int kc     = K >> 2;                   // float4 chunks per row
    const int chunks = 16 * kc;                  // 512 (K=128) or 256 (K=64)
    const float* Ablk = A + (size_t)tm * 16 * K; // stripe is contiguous
    for (int c = tid; c < chunks; c += 128) {    // uniform trip count
        int row  = c / kc;
        int col4 = (c - row * kc) << 2;
        int lidx = row * ldsK + col4;
#if USE_ASYNC_LDS
        __builtin_amdgcn_global_load_async_to_lds_b128(
            (v4i*)(Ablk + row * K + col4), (v4i*)&lds_a[lidx], 0, 0);
#else
        const float4 v = *(const float4*)(Ablk + row * K + col4);
        *(float4*)&lds_a[lidx] = v;
#endif
    }
#if USE_ASYNC_LDS
#if __has_builtin(__builtin_amdgcn_s_wait_asynccnt)
    __builtin_amdgcn_s_wait_asynccnt(0);
#else
    asm volatile("s_wait_asynccnt 0x0" ::: "memory");
#endif
#endif
    __syncthreads();

    // ---- WMMA K loop ----
    v8f acc = {};
    for (int k0 = 0; k0 < K; k0 += 4) {
        // A 16x4 layout (ISA 7.12.2): lanes 0-15 K={0,1}, lanes 16-31 K={2,3}
        v2f a, b;
        a.x = lds_a[m * ldsK + k0 + khalf];
        a.y = lds_a[m * ldsK + k0 + khalf + 1];
        // B 4x16: row striped across lanes; lane halves select K pair
        b.x = W[(k0 + khalf) * HID + n];
        b.y = W[(k0 + khalf + 1) * HID + n];
        acc = __builtin_amdgcn_wmma_f32_16x16x4_f32(
            /*neg_a=*/false, a, /*neg_b=*/false, b,
            /*c_mod=*/(short)0, acc, /*reuse_a=*/false, /*reuse_b=*/false);
    }
    // C/D layout: VGPR v -> row m = v + 8*half, col n
    const int rbase = tm * 16 + (half << 3);
#pragma unroll
    for (int v = 0; v < 8; ++v) {
        C[(rbase + v) * HID + n] = acc[v];
    }
}

// ---------------- aggregation: out = scatter_add(norm * t[src]) + bias ----------------
// Self-loop (src=dst=i, norm=dis[i]^2) + bias folded into the init pass.
__global__ void agg_init_kernel(const float* __restrict__ t,
                                const float* __restrict__ dis,
                                const float* __restrict__ bias,
                                float* __restrict__ out, int n64) {
    int i = blockIdx.x * blockDim.x + threadIdx.x;
    if (i >= n64) return;
    int node = i >> 6;
    int f    = i & 63;
    float d  = dis[node];
    out[i] = t[i] * d * d + bias[f];
}

// One wave per edge; lane L handles features L and L+32 (coalesced 256B per edge).
__global__ void agg_edges_kernel(const float* __restrict__ t,
                                 const int* __restrict__ src,
                                 const int* __restrict__ dst,
                                 const float* __restrict__ dis,
                                 float* __restrict__ out, int e) {
    int gid  = blockIdx.x * blockDim.x + threadIdx.x;
    int edge = gid >> 5;
    int lane = gid & 31;
    if (edge >= e) return;
    int s = src[edge];
    int d = dst[edge];
    float nrm = dis[s] * dis[d];
    float v0 = t[s * HID + lane]      * nrm;
    float v1 = t[s * HID + lane + 32] * nrm;
    atomicAdd(&out[d * HID + lane],      v0);
    atomicAdd(&out[d * HID + lane + 32], v1);
}

__global__ void relu_kernel(float* h, int n) {
    int i = blockIdx.x * blockDim.x + threadIdx.x;
    if (i < n) h[i] = fmaxf(h[i], 0.0f);
}

// ---------------- global mean pool ----------------
__global__ void zero_kernel(float* p, int n) {
    int i = blockIdx.x * blockDim.x + threadIdx.x;
    if (i < n) p[i] = 0.0f;
}

__global__ void pool_accum_kernel(const float* __restrict__ h,
                                  const int* __restrict__ batch,
                                  float* __restrict__ sums, int n64) {
    int i = blockIdx.x * blockDim.x + threadIdx.x;
    if (i >= n64) return;
    int node = i >> 6;
    int f    = i & 63;
    atomicAdd(&sums[batch[node] * HID + f], h[i]);
}

__global__ void pool_count_kernel(const int* __restrict__ batch,
                                  float* __restrict__ cnts, int n) {
    int i = blockIdx.x * blockDim.x + threadIdx.x;
    if (i < n) atomicAdd(&cnts[batch[i]], 1.0f);
}

// ---------------- classifier: relu(pooled/cnt @ Wc1 + bc1) @ Wc2 + bc2 ----------------
// Single block, thread g owns graph g. Tiny (64x64x32 + 64x32x2).
__global__ void classifier_kernel(const float* __restrict__ pooled,
                                  const float* __restrict__ cnts,
                                  const float* __restrict__ Wc1,
                                  const float* __restrict__ bc1,
                                  const float* __restrict__ Wc2,
                                  const float* __restrict__ bc2,
                                  float* __restrict__ out) {
    int g = threadIdx.x;
    if (g >= 64) return;
    float inv = 1.0f / fmaxf(cnts[g], 1.0f);
    float row[HID];
#pragma unroll
    for (int f = 0; f < HID; ++f) row[f] = pooled[g * HID + f] * inv;
    float hc[32];
    for (int j = 0; j < 32; ++j) {
        float s = bc1[j];
#pragma unroll
        for (int f = 0; f < HID; ++f) s += row[f] * Wc1[f * 32 + j];
        hc[j] = fmaxf(s, 0.0f);
    }
    for (int o = 0; o < 2; ++o) {
        float s = bc2[o];
#pragma unroll
        for (int j = 0; j < 32; ++j) s += hc[j] * Wc2[j * 2 + o];
        out[g * 2 + o] = s;
    }
}

extern "C" void kernel_launch(void* const* d_in, const int* in_sizes, int n_in,
                              void* d_out, int out_size, void* d_ws, size_t ws_size,
                              hipStream_t stream) {
    const float* x   = (const float*)d_in[0];
    const int*   ei  = (const int*)d_in[1];      // [2, E]: row 0 = src, row 1 = dst
    const int*   bat = (const int*)d_in[2];
    const float* W1  = (const float*)d_in[3];
    const float* b1  = (const float*)d_in[4];
    const float* W2  = (const float*)d_in[5];
    const float* b2  = (const float*)d_in[6];
    const float* W3  = (const float*)d_in[7];
    const float* b3  = (const float*)d_in[8];
    const float* Wc1 = (const float*)d_in[9];
    const float* bc1 = (const float*)d_in[10];
    const float* Wc2 = (const float*)d_in[11];
    const float* bc2 = (const float*)d_in[12];
    float* out = (float*)d_out;

    const int n = in_sizes[2];          // 100000 nodes
    const int e = in_sizes[1] / 2;      // 1600000 edges
    const int* src = ei;
    const int* dst = ei + e;

    // ---- workspace partition (all f32) ----
    float* dis    = (float*)d_ws;           // [n]      degree -> dis
    float* bufA   = dis + n;                // [n*HID]  GEMM output t = h @ W
    float* bufB   = bufA + (size_t)n * HID; // [n*HID]  aggregated layer output
    float* pooled = bufB + (size_t)n * HID; // [64*HID]
    float* cnts   = pooled + 64 * HID;      // [64]

    const int B = 256;

    // degree + symmetric normalization
    init_deg_kernel<<<(n + B - 1) / B, B, 0, stream>>>(dis, n);
    accum_deg_kernel<<<(e + B - 1) / B, B, 0, stream>>>(dst, dis, e);
    rsqrt_deg_kernel<<<(n + B - 1) / B, B, 0, stream>>>(dis, n);

    const int n64       = n * HID;
    const int gemm_grid = n / 16;           // 6250 row tiles; 4 waves cover N=64
    const int agg_grid  = (e * 32 + B - 1) / B;
    const int el_grid   = (n64 + B - 1) / B;

    // ---- layer 1: 128 -> 64, relu ----
    gemm_wmma_f32<<<gemm_grid, 128, 0, stream>>>(x, W1, bufA, 128);
    agg_init_kernel<<<el_grid, B, 0, stream>>>(bufA, dis, b1, bufB, n64);
    agg_edges_kernel<<<agg_grid, B, 0, stream>>>(bufA, src, dst, dis, bufB, e);
    relu_kernel<<<el_grid, B, 0, stream>>>(bufB, n64);

    // ---- layer 2: 64 -> 64, relu ----
    gemm_wmma_f32<<<gemm_grid, 128, 0, stream>>>(bufB, W2, bufA, HID);
    agg_init_kernel<<<el_grid, B, 0, stream>>>(bufA, dis, b2, bufB, n64);
    agg_edges_kernel<<<agg_grid, B, 0, stream>>>(bufA, src, dst, dis, bufB, e);
    relu_kernel<<<el_grid, B, 0, stream>>>(bufB, n64);

    // ---- layer 3: 64 -> 64 (no relu) ----
    gemm_wmma_f32<<<gemm_grid, 128, 0, stream>>>(bufB, W3, bufA, HID);
    agg_init_kernel<<<el_grid, B, 0, stream>>>(bufA, dis, b3, bufB, n64);
    agg_edges_kernel<<<agg_grid, B, 0, stream>>>(bufA, src, dst, dis, bufB, e);

    // ---- global mean pool ----
    zero_kernel<<<(64 * HID + 64 + B - 1) / B, B, 0, stream>>>(pooled, 64 * HID + 64);
    pool_accum_kernel<<<el_grid, B, 0, stream>>>(bufB, bat, pooled, n64);
    pool_count_kernel<<<(n + B - 1) / B, B, 0, stream>>>(bat, cnts, n);

    // ---- classifier ----
    classifier_kernel<<<1, 64, 0, stream>>>(pooled, cnts, Wc1, bc1, Wc2, bc2, out);
}